// Arb_upsample_48223892800153
// MI455X (gfx1250) — compile-verified
//
#include <hip/hip_runtime.h>

typedef __attribute__((ext_vector_type(2))) float v2f;
typedef __attribute__((ext_vector_type(8))) float v8f;

// ---------------------------------------------------------------------------
// Implicit-GEMM convolution on the CDNA5 matrix pipe (V_WMMA_F32_16X16X4_F32).
//   D(M = N*H*W pixels x N = Co) = im2col(A) * W^T,  K = Ci*ks*ks.
// K is enumerated tap-major ((kh,kw) outer, ci inner) — identical sum, but all
// div/mod and pad logic hoists out of the hot loop: the ci loop is two
// stride-H*W input loads + two stride-ks^2 weight loads + one WMMA.
// Zero padding is branchless (clamped address * mask) so EXEC is all-ones at
// every WMMA (ISA 7.12 requirement).
// f32 WMMA operand layout (ISA 7.12.2):
//   A 16x4 : lane(0-15)=row M, vgpr0/1 = K0/K1 ; lanes 16-31 = K2/K3
//   B 4x16 : lane(0-15)=col N, vgpr0/1 = K0/K1 ; lanes 16-31 = K2/K3
//   D 16x16: lane=col N (mod 16), vgpr v = row M (v, or v+8 for lanes 16-31)
// Epilogue fuses bias, BatchNorm affine, ReLU/sigmoid, optional pixel-shuffle.
// ---------------------------------------------------------------------------
__global__ __launch_bounds__(128) void conv_wmma_kernel(
    const float* __restrict__ in, const float* __restrict__ wt,
    const float* __restrict__ bias,
    const float* __restrict__ bng, const float* __restrict__ bnb,
    const float* __restrict__ bnm, const float* __restrict__ bnv,
    float* __restrict__ out,
    int N, int Ci, int H, int W, int Co, int ks, int act, int shuf_r)
{
  const int ksq = ks * ks;
  const int pad = ks >> 1;
  const size_t HW = (size_t)H * W;
  const int M = N * H * W;
  const int mtiles = (M + 15) >> 4;
  const int ctiles = (Co + 15) >> 4;
  const long tiles = (long)mtiles * ctiles;
  long wave = (long)blockIdx.x * (blockDim.x >> 5) + (threadIdx.x >> 5);
  if (wave >= tiles) return;               // wave-uniform: EXEC stays all-ones
  const int lane   = threadIdx.x & 31;
  const int tile_m = (int)(wave % mtiles) * 16;
  const int tile_c = (int)(wave / mtiles) * 16;
  const int half   = lane >> 4;            // 0: K pair {0,1}, 1: K pair {2,3}
  const int l      = lane & 15;

  // A-side pixel for this lane (clamped; stores are masked)
  int m  = tile_m + l;
  int mm = (m < M) ? m : (M - 1);
  int n   = mm / (H * W);
  int rem = mm - n * H * W;
  int hh  = rem / W;
  int ww  = rem - hh * W;
  // B-side output channel for this lane
  int co  = tile_c + l;
  int cob = (co < Co) ? co : (Co - 1);
  const float* __restrict__ wrow = wt + (size_t)cob * (size_t)Ci * ksq;
  const float* __restrict__ inn  = in + (size_t)n * Ci * HW;

  v8f acc = {};
  for (int kh = 0; kh < ks; ++kh) {
    const int ih  = hh + kh - pad;
    const int ihc = (ih < 0) ? 0 : ((ih > H - 1) ? (H - 1) : ih);
    for (int kw = 0; kw < ks; ++kw) {
      const int iw  = ww + kw - pad;
      const int iwc = (iw < 0) ? 0 : ((iw > W - 1) ? (W - 1) : iw);
      const float vmask =
          (ih >= 0 && ih < H && iw >= 0 && iw < W) ? 1.f : 0.f;
      // lane's first ci of each K-group of 4 is cb + half*2
      const float* pa = inn + (size_t)ihc * W + iwc + (size_t)(half * 2) * HW;
      const float* pb = wrow + (kh * ks + kw) + (half * 2) * ksq;
      __builtin_prefetch(pb + 8 * ksq, 0, 3);   // global_prefetch_b8
#pragma unroll 4
      for (int cb = 0; cb < Ci; cb += 4) {
        v2f A = { pa[0] * vmask, pa[HW] * vmask };
        v2f B = { pb[0], pb[ksq] };
        acc = __builtin_amdgcn_wmma_f32_16x16x4_f32(false, A, false, B,
                                                    (short)0, acc, false, false);
        pa += 4 * HW;
        pb += 4 * ksq;
      }
    }
  }

  // epilogue: this lane owns column (out channel) cod for 8 pixel rows
  int cod = tile_c + l;
  if (cod < Co) {
    float s = 1.f, sh = 0.f;
    if (bng) {
      float sc = bng[cod] * rsqrtf(bnv[cod] + 1e-5f);
      s = sc; sh = bnb[cod] - bnm[cod] * sc;
    }
    float bs = bias ? bias[cod] : 0.f;
#pragma unroll
    for (int v = 0; v < 8; ++v) {
      int md = tile_m + v + half * 8;
      if (md < M) {
        float val = acc[v] + bs;
        val = val * s + sh;
        if (act == 1)      val = fmaxf(val, 0.f);
        else if (act == 2) val = 1.f / (1.f + expf(-val));
        int n2 = md / (H * W);
        int r2 = md - n2 * H * W;
        int h2 = r2 / W;
        int w2 = r2 - h2 * W;
        if (shuf_r <= 1) {
          out[(((size_t)n2 * Co + cod) * H + h2) * W + w2] = val;
        } else {
          int rr2 = shuf_r * shuf_r;
          int Cout = Co / rr2;
          int c2 = cod / rr2;
          int t2 = cod - c2 * rr2;
          int ii = t2 / shuf_r;
          int jj = t2 - ii * shuf_r;
          int Ho = H * shuf_r, Wo = W * shuf_r;
          out[(((size_t)n2 * Cout + c2) * Ho + (h2 * shuf_r + ii)) * Wo +
              (w2 * shuf_r + jj)] = val;
        }
      }
    }
  }
}

// ---------------- routing MLP: rout = softmax(relu([1/s,1/s2] W1^T+b1) W2^T+b2)
__global__ void routing_kernel(const int* __restrict__ scale, const int* __restrict__ scale2,
                               const float* __restrict__ r1w, const float* __restrict__ r1b,
                               const float* __restrict__ r2w, const float* __restrict__ r2b,
                               float* __restrict__ rout)
{
  __shared__ float hdn[64];
  int t = threadIdx.x;
  float r0 = 1.f / (float)(*scale);
  float r1 = 1.f / (float)(*scale2);
  if (t < 64) hdn[t] = fmaxf(r0 * r1w[t * 2 + 0] + r1 * r1w[t * 2 + 1] + r1b[t], 0.f);
  __syncthreads();
  if (t == 0) {
    float l0 = r2b[0], l1 = r2b[1];
    for (int j = 0; j < 64; ++j) { l0 += r2w[j] * hdn[j]; l1 += r2w[64 + j] * hdn[j]; }
    float mx = fmaxf(l0, l1);
    float e0 = expf(l0 - mx), e1 = expf(l1 - mx);
    float inv = 1.f / (e0 + e1);
    rout[0] = e0 * inv; rout[1] = e1 * inv;
  }
}

// ---------------- blend expert weight pool: w = r0*wp[0] + r1*wp[1]
__global__ void blend_kernel(const float* __restrict__ wp, const float* __restrict__ rout,
                             float* __restrict__ w, int E)
{
  long i = (long)blockIdx.x * blockDim.x + threadIdx.x;
  if (i < E) w[i] = rout[0] * wp[i] + rout[1] * wp[E + i];
}

// ---------------- 2x2 average pool
__global__ void avgpool2_kernel(const float* __restrict__ in, float* __restrict__ out,
                                int N, int C, int H, int W)
{
  int oh = H >> 1, ow = W >> 1;
  long idx = (long)blockIdx.x * blockDim.x + threadIdx.x;
  long total = (long)N * C * oh * ow;
  if (idx >= total) return;
  int ox = idx % ow; long t = idx / ow;
  int oy = t % oh; t /= oh;
  int c = t % C; int n = (int)(t / C);
  const float* p = in + ((size_t)n * C + c) * H * W;
  out[idx] = 0.25f * (p[(2 * oy) * W + 2 * ox]     + p[(2 * oy) * W + 2 * ox + 1] +
                      p[(2 * oy + 1) * W + 2 * ox] + p[(2 * oy + 1) * W + 2 * ox + 1]);
}

// ---------------- generic bilinear resize (align_corners=False rule)
__global__ void resize_bilinear_kernel(const float* __restrict__ in, float* __restrict__ out,
                                       int N, int C, int h, int w, int oh, int ow)
{
  long idx = (long)blockIdx.x * blockDim.x + threadIdx.x;
  long total = (long)N * C * oh * ow;
  if (idx >= total) return;
  int ox = idx % ow; long t = idx / ow;
  int oy = t % oh; t /= oh;
  int c = t % C; int n = (int)(t / C);
  float cy = fmaxf((oy + 0.5f) * ((float)h / oh) - 0.5f, 0.f);
  int y0 = (int)floorf(cy); y0 = (y0 > h - 1) ? (h - 1) : y0;
  int y1 = (y0 + 1 < h - 1) ? (y0 + 1) : (h - 1);
  float ty = cy - (float)y0;
  float cx = fmaxf((ox + 0.5f) * ((float)w / ow) - 0.5f, 0.f);
  int x0 = (int)floorf(cx); x0 = (x0 > w - 1) ? (w - 1) : x0;
  int x1 = (x0 + 1 < w - 1) ? (x0 + 1) : (w - 1);
  float tx = cx - (float)x0;
  const float* p = in + ((size_t)n * C + c) * h * w;
  float a = p[y0 * w + x0] * (1.f - ty) + p[y1 * w + x0] * ty;
  float b = p[y0 * w + x1] * (1.f - ty) + p[y1 * w + x1] * ty;
  out[idx] = a * (1.f - tx) + b * tx;
}

// ---------------- out = x + sc * mask (mask broadcast over channels)
__global__ void fuse_sa_kernel(const float* __restrict__ xin, const float* __restrict__ sc,
                               const float* __restrict__ mask, float* __restrict__ out,
                               int N, int C, int HW)
{
  long idx = (long)blockIdx.x * blockDim.x + threadIdx.x;
  long total = (long)N * C * HW;
  if (idx >= total) return;
  int p = idx % HW; long t = idx / HW;
  int n = (int)(t / C);
  out[idx] = xin[idx] + sc[idx] * mask[(size_t)n * HW + p];
}

// ---------------- offset field: analytic 4-ch grid -> 1x1 MLP (4->32->32->2)
// Writes RAW offsets; grid normalization cancels in the sampler.
__global__ __launch_bounds__(256) void offset_field_kernel(
    const int* __restrict__ scale, const int* __restrict__ scale2,
    const float* __restrict__ b1w, const float* __restrict__ b1b,
    const float* __restrict__ b2w, const float* __restrict__ b2b,
    const float* __restrict__ oww, const float* __restrict__ obb,
    float* __restrict__ off, int r, int Hc, int Wc)
{
  __shared__ float sW1[128], sB1[32], sW2[1024], sB2[32], sWo[64], sBo[2];
  for (int i = threadIdx.x; i < 128;  i += blockDim.x) sW1[i] = b1w[i];
  for (int i = threadIdx.x; i < 32;   i += blockDim.x) sB1[i] = b1b[i];
  for (int i = threadIdx.x; i < 1024; i += blockDim.x) sW2[i] = b2w[i];
  for (int i = threadIdx.x; i < 32;   i += blockDim.x) sB2[i] = b2b[i];
  for (int i = threadIdx.x; i < 64;   i += blockDim.x) sWo[i] = oww[i];
  if (threadIdx.x < 2) sBo[threadIdx.x] = obb[threadIdx.x];
  __syncthreads();
  long idx = (long)blockIdx.x * blockDim.x + threadIdx.x;
  if (idx >= (long)Hc * Wc) return;
  int ox = idx % Wc, oy = (int)(idx / Wc);
  float sh = (float)(*scale)  / (float)r;
  float sw = (float)(*scale2) / (float)r;
  float chv = (oy + 0.5f) / sh; float coh = chv - floorf(chv + 0.001f) - 0.5f;
  float cwv = (ox + 0.5f) / sw; float cow = cwv - floorf(cwv + 0.001f) - 0.5f;
  float i0v = 1.f / sw, i1v = 1.f / sh;
  float e1[32];
#pragma unroll
  for (int o = 0; o < 32; ++o)
    e1[o] = fmaxf(sW1[o * 4] * i0v + sW1[o * 4 + 1] * i1v +
                  sW1[o * 4 + 2] * coh + sW1[o * 4 + 3] * cow + sB1[o], 0.f);
  float o0 = sBo[0], o1 = sBo[1];
  for (int o = 0; o < 32; ++o) {
    float a = sB2[o];
#pragma unroll
    for (int j = 0; j < 32; ++j) a += sW2[o * 32 + j] * e1[j];
    float e2 = fmaxf(a, 0.f);
    o0 += sWo[o] * e2;
    o1 += sWo[32 + o] * e2;
  }
  off[idx] = o0;                       // X offset (channel 0)
  off[(long)Hc * Wc + idx] = o1;       // Y offset (channel 1)
}

// ---------------- zero-padded bilinear grid sample, interleaved-channel store
// fx = (ox+0.5)/sw - 0.5 + off0 ; fy = (oy+0.5)/sh - 0.5 + off1
__global__ void grid_sample_kernel(const float* __restrict__ src, const float* __restrict__ off,
                                   float* __restrict__ inter,
                                   const int* __restrict__ scale, const int* __restrict__ scale2,
                                   int N, int C, int hin, int win, int Hc, int Wc,
                                   int lvl, int r, int Cinter)
{
  long idx = (long)blockIdx.x * blockDim.x + threadIdx.x;
  long total = (long)N * C * Hc * Wc;
  if (idx >= total) return;
  int ox = idx % Wc; long t = idx / Wc;
  int oy = t % Hc; t /= Hc;
  int c = t % C; int n = (int)(t / C);
  float sh = (float)(*scale)  / (float)r;
  float sw = (float)(*scale2) / (float)r;
  float fx = (ox + 0.5f) / sw - 0.5f + off[(long)oy * Wc + ox];
  float fy = (oy + 0.5f) / sh - 0.5f + off[(long)Hc * Wc + (long)oy * Wc + ox];
  float x0f = floorf(fx), y0f = floorf(fy);
  float tx = fx - x0f, ty = fy - y0f;
  int x0 = (int)x0f, y0 = (int)y0f;
  const float* p = src + ((size_t)n * C + c) * hin * win;
  float v00 = (x0 >= 0 && x0 < win && y0 >= 0 && y0 < hin)         ? p[y0 * win + x0]             : 0.f;
  float v01 = (x0 + 1 >= 0 && x0 + 1 < win && y0 >= 0 && y0 < hin) ? p[y0 * win + x0 + 1]         : 0.f;
  float v10 = (x0 >= 0 && x0 < win && y0 + 1 >= 0 && y0 + 1 < hin) ? p[(y0 + 1) * win + x0]       : 0.f;
  float v11 = (x0 + 1 >= 0 && x0 + 1 < win && y0 + 1 >= 0 && y0 + 1 < hin) ? p[(y0 + 1) * win + x0 + 1] : 0.f;
  float v = v00 * (1.f - tx) * (1.f - ty) + v01 * tx * (1.f - ty) +
            v10 * (1.f - tx) * ty + v11 * tx * ty;
  inter[(((size_t)n * Cinter + (c * 3 + lvl)) * Hc + oy) * Wc + ox] = v;
}

// ---------------- grouped 1x1 conv (96->32, groups=32) + ReLU
__global__ void gconv_relu_kernel(const float* __restrict__ in, const float* __restrict__ w,
                                  const float* __restrict__ b, float* __restrict__ out,
                                  int N, int HW)
{
  long idx = (long)blockIdx.x * blockDim.x + threadIdx.x;
  long total = (long)N * 32 * HW;
  if (idx >= total) return;
  int p = idx % HW; long t = idx / HW;
  int o = t % 32; int n = (int)(t / 32);
  const float* base = in + ((size_t)n * 96 + 3 * o) * HW + p;
  float v = b[o] + w[o * 3] * base[0] + w[o * 3 + 1] * base[(size_t)HW] +
            w[o * 3 + 2] * base[2 * (size_t)HW];
  out[idx] = fmaxf(v, 0.f);
}

// ---------------------------------------------------------------------------
extern "C" void kernel_launch(void* const* d_in, const int* in_sizes, int n_in,
                              void* d_out, int out_size, void* d_ws, size_t ws_size,
                              hipStream_t stream)
{
  (void)in_sizes; (void)n_in; (void)out_size;
  // setup_inputs() recursive dict-insertion-order flattening:
  const float* x      = (const float*)d_in[0];
  const int*   scale  = (const int*)d_in[1];
  const int*   scale2 = (const int*)d_in[2];
  // sa_adapt param layout (29 leaves)
  enum { I_M1W = 0, I_M1B, I_BN1G, I_BN1B, I_BN1M, I_BN1V,
         I_M2W, I_M2B, I_BN2G, I_BN2B, I_BN2M, I_BN2V,
         I_M3W, I_M3B, I_BN3G, I_BN3B, I_BN3M, I_BN3V,
         I_M4W, I_M4B, I_BN4G, I_BN4B, I_BN4M, I_BN4V,
         I_R1W, I_R1B, I_R2W, I_R2B, I_WPOOL };
  const float* P64[29]; for (int j = 0; j < 29; ++j) P64[j] = (const float*)d_in[3 + j];
  const float* P32[29]; for (int j = 0; j < 29; ++j) P32[j] = (const float*)d_in[32 + j];
  const float* psw[3] = { (const float*)d_in[61], (const float*)d_in[63], (const float*)d_in[65] };
  const float* psb[3] = { (const float*)d_in[62], (const float*)d_in[64], (const float*)d_in[66] };
  const float* b1w = (const float*)d_in[67]; const float* b1b = (const float*)d_in[68];
  const float* b2w = (const float*)d_in[69]; const float* b2b = (const float*)d_in[70];
  const float* oww = (const float*)d_in[71]; const float* obb = (const float*)d_in[72];
  const float* gaw = (const float*)d_in[73]; const float* gab = (const float*)d_in[74];
  const float* gbw = (const float*)d_in[75]; const float* gbb = (const float*)d_in[76];
  const float* redw = (const float*)d_in[77]; const float* redb = (const float*)d_in[78];
  float* outp = (float*)d_out;

  // ---- workspace layout (floats); aggressive aliasing keeps it ~329 MB ----
  if (ws_size < (size_t)82197568 * 4) return;
  float* ws = (float*)d_ws;
  float* fea    = ws + 0;          // 2*64*96*96  = 1179648
  float* m196   = ws + 1179648;    // 2*32*96*96  =  589824
  float* pool48 = ws + 1769472;    //               147456
  float* m248   = ws + 1916928;    //               147456
  float* m348   = ws + 2064384;    //               147456
  float* up96   = ws + 2211840;    //               589824
  float* mask96 = ws + 2801664;    //                18432
  float* w64    = ws + 2820096;    // 64*64*9     =  36864
  float* w32    = ws + 2856960;    // 32*32*9     =   9216
  float* rout   = ws + 2866176;    //                   64
  float* offb   = ws + 2866240;    // 2*384*384   =  294912
  float* sc96   = ws + 3161152;    // 2*64*96*96  = 1179648
  float* inter  = ws + 4340800;    // 2*96*384*384= 28311552
  float* shuf0  = ws + 32652352;   // 2*32*192^2  =  2359296
  float* shuf1  = ws + 35011648;   // 2*32*384^2  =  9437184
  float* shuf2  = ws + 44448832;   // 2*32*768^2  = 37748736  (total 82197568)
  // aliases valid by liveness:
  float* bufA    = shuf1;             // free after level-1 sampling
  float* bufB    = shuf2 + 0;         // free after level-2 sampling
  float* m1_384  = inter;             // free after grouped conv
  float* pool192 = shuf0;             // free after level-0 sampling
  float* m2_192  = shuf2 + 9437184;
  float* m3_192  = shuf2 + 11796480;
  float* up384   = shuf2 + 14155776;
  float* mask384 = shuf2 + 23592960;
  float* sc384   = shuf2 + 23887872;  // ends at 33325056 < 37748736

  auto conv = [&](const float* in, const float* wt, const float* bias,
                  const float* bg, const float* bb, const float* bm, const float* bv,
                  float* out, int N, int Ci, int H, int W, int Co, int ks, int act, int shufr) {
    int M = N * H * W;
    long tiles = (long)((M + 15) / 16) * ((Co + 15) / 16);
    int blocks = (int)((tiles + 3) / 4);     // 4 waves (128 threads) per block
    conv_wmma_kernel<<<dim3(blocks), dim3(128), 0, stream>>>(
        in, wt, bias, bg, bb, bm, bv, out, N, Ci, H, W, Co, ks, act, shufr);
  };

  auto run_sa_adapt = [&](const float* const* P, const float* xin, float* xout,
                          int C, int Hh, int Ww,
                          float* m1buf, float* poolbuf, float* m2buf, float* m3buf,
                          float* upbuf, float* maskbuf, float* scbuf,
                          float* wblend, float* routp) {
    routing_kernel<<<1, 64, 0, stream>>>(scale, scale2, P[I_R1W], P[I_R1B], P[I_R2W], P[I_R2B], routp);
    int E = C * C * 9;
    blend_kernel<<<(E + 255) / 256, 256, 0, stream>>>(P[I_WPOOL], routp, wblend, E);
    conv(xin, P[I_M1W], P[I_M1B], P[I_BN1G], P[I_BN1B], P[I_BN1M], P[I_BN1V],
         m1buf, 2, C, Hh, Ww, 32, 3, /*relu*/1, 0);
    { long tot = (long)2 * 32 * (Hh / 2) * (Ww / 2);
      avgpool2_kernel<<<(int)((tot + 255) / 256), 256, 0, stream>>>(m1buf, poolbuf, 2, 32, Hh, Ww); }
    conv(poolbuf, P[I_M2W], P[I_M2B], P[I_BN2G], P[I_BN2B], P[I_BN2M], P[I_BN2V],
         m2buf, 2, 32, Hh / 2, Ww / 2, 32, 3, 1, 0);
    conv(m2buf, P[I_M3W], P[I_M3B], P[I_BN3G], P[I_BN3B], P[I_BN3M], P[I_BN3V],
         m3buf, 2, 32, Hh / 2, Ww / 2, 32, 3, 1, 0);
    { long tot = (long)2 * 32 * Hh * Ww;
      resize_bilinear_kernel<<<(int)((tot + 255) / 256), 256, 0, stream>>>(
          m3buf, upbuf, 2, 32, Hh / 2, Ww / 2, Hh, Ww); }
    conv(upbuf, P[I_M4W], P[I_M4B], P[I_BN4G], P[I_BN4B], P[I_BN4M], P[I_BN4V],
         maskbuf, 2, 32, Hh, Ww, 1, 3, /*sigmoid*/2, 0);
    conv(xin, wblend, nullptr, nullptr, nullptr, nullptr, nullptr,
         scbuf, 2, C, Hh, Ww, C, 3, 0, 0);
    { long tot = (long)2 * C * Hh * Ww;
      fuse_sa_kernel<<<(int)((tot + 255) / 256), 256, 0, stream>>>(
          xin, scbuf, maskbuf, xout, 2, C, Hh * Ww); }
  };

  // ---- sa_adapt on the 64-channel input -> fea -------------------------------
  run_sa_adapt(P64, x, fea, 64, 96, 96, m196, pool48, m248, m348, up96, mask96,
               sc96, w64, rout + 0);

  // ---- three pixel-shuffle branches + learned-offset resample to 384x384 -----
  float* shufb[3] = { shuf0, shuf1, shuf2 };
  for (int lvl = 0; lvl < 3; ++lvl) {
    int r = 2 << lvl;                               // 2, 4, 8
    conv(fea, psw[lvl], psb[lvl], nullptr, nullptr, nullptr, nullptr,
         shufb[lvl], 2, 64, 96, 96, 32 * r * r, 3, 0, /*pixel shuffle*/r);
    { long tot = (long)384 * 384;
      offset_field_kernel<<<(int)((tot + 255) / 256), 256, 0, stream>>>(
          scale, scale2, b1w, b1b, b2w, b2b, oww, obb, offb, r, 384, 384); }
    { long tot = (long)2 * 32 * 384 * 384;
      grid_sample_kernel<<<(int)((tot + 255) / 256), 256, 0, stream>>>(
          shufb[lvl], offb, inter, scale, scale2, 2, 32, 96 * r, 96 * r,
          384, 384, lvl, r, 96); }
  }

  // ---- grouped 1x1 (96->32, g=32) + ReLU, then 1x1 (32->32) + ReLU -----------
  { long tot = (long)2 * 32 * 384 * 384;
    gconv_relu_kernel<<<(int)((tot + 255) / 256), 256, 0, stream>>>(
        inter, gaw, gab, bufA, 2, 384 * 384); }
  conv(bufA, gbw, gbb, nullptr, nullptr, nullptr, nullptr,
       bufB, 2, 32, 384, 384, 32, 1, 1, 0);

  // ---- sa_adapt on 32 channels at 384x384 (in-place on bufB) -----------------
  run_sa_adapt(P32, bufB, bufB, 32, 384, 384, m1_384, pool192, m2_192, m3_192,
               up384, mask384, sc384, w32, rout + 2);

  // ---- final 3x3 conv to RGB -> d_out ----------------------------------------
  conv(bufB, redw, redb, nullptr, nullptr, nullptr, nullptr,
       outp, 2, 32, 384, 384, 3, 3, 0, 0);
}